// EdgeEmbedding_74526272520544
// MI455X (gfx1250) — compile-verified
//
#include <hip/hip_runtime.h>
#include <hip/hip_bf16.h>
#include <stdint.h>

#define RCINV_F 0.2f
#define PI_F 3.1415927f
#define BASIS 64
#define BLOCK 128
#define EDGES_PER_TILE 128
#define PITCH 68   // floats per LDS row: 64 + 4 pad -> conflict-free b128 (bank step 4/lane)

typedef float v4f __attribute__((ext_vector_type(4)));

__global__ __launch_bounds__(BLOCK) void edge_embed_kernel(
    const float* __restrict__ len, float* __restrict__ out, int n)
{
    __shared__ __align__(16) float tile[EDGES_PER_TILE * PITCH];  // 34,816 B

    const int tid = threadIdx.x;
    const long long blockBase = (long long)blockIdx.x * EDGES_PER_TILE;
    const long long edge = blockBase + tid;

    float d = 0.0f;
    if (edge < (long long)n) d = len[edge];

    // fc = 0.5*cos(pi*d/rc) + 0.5 ; scaled output o_m = (T_m + 1)*0.5*fc = fma(T_m, h, h)
    const float fc = 0.5f * cosf(PI_F * RCINV_F * d) + 0.5f;
    const float h  = 0.5f * fc;
    const float u  = d * RCINV_F - 1.0f;
    const float x  = 2.0f * u * u - 1.0f;
    const float x2 = 2.0f * x;

    float* row = &tile[tid * PITCH];

    // Chebyshev recurrence T_m = 2x*T_{m-1} - T_{m-2}, packed 4-wide into LDS (ds_store_b128)
    float tm2 = 1.0f;   // T0
    float tm1 = x;      // T1
    {
        float a = tm2;            // T0
        float b = tm1;            // T1
        float c = x2 * b - a;     // T2
        float e = x2 * c - b;     // T3
        v4f q = { fmaf(a,h,h), fmaf(b,h,h), fmaf(c,h,h), fmaf(e,h,h) };
        *reinterpret_cast<v4f*>(row) = q;
        tm2 = c; tm1 = e;
    }
#pragma unroll
    for (int g = 1; g < 16; ++g) {
        float a = x2 * tm1 - tm2;
        float b = x2 * a  - tm1;
        float c = x2 * b  - a;
        float e = x2 * c  - b;
        v4f q = { fmaf(a,h,h), fmaf(b,h,h), fmaf(c,h,h), fmaf(e,h,h) };
        *reinterpret_cast<v4f*>(row + 4*g) = q;
        tm2 = c; tm1 = e;
    }

    __syncthreads();

    // Coalesced LDS -> global via CDNA5 async store-from-LDS (ASYNCcnt path).
    // Iter i, lane t handles flat float4 (i*128 + t):
    //   lds   = base + (t>>4)*272 + (t&15)*16 + i*2176
    //   gaddr = out  + blockBase*256B + t*16 + i*2048
    // -> consecutive lanes write consecutive 16B chunks: one fully-coalesced
    //    512B global write per wave-instruction.
    const unsigned ldsBase = (unsigned)(unsigned long long)(uintptr_t)&tile[0];
    long long validE = (long long)n - blockBase;
    if (validE > EDGES_PER_TILE) validE = EDGES_PER_TILE;

    unsigned lds = ldsBase + (unsigned)((tid >> 4) * (PITCH * 4) + (tid & 15) * 16);
    unsigned long long gaddr =
        (unsigned long long)(uintptr_t)(out + blockBase * BASIS) +
        (unsigned long long)tid * 16ull;

    if (validE == EDGES_PER_TILE) {
        // Hot path: full tile, uniform EXEC, no per-store guards.
#pragma unroll
        for (int i = 0; i < 16; ++i) {
            asm volatile("global_store_async_from_lds_b128 %0, %1, off"
                         :: "v"(gaddr), "v"(lds) : "memory");
            lds   += 8u * (PITCH * 4);   // 8 edge rows per iteration (2176 B)
            gaddr += (unsigned long long)(BLOCK * 16);  // 2048 B
        }
    } else if (validE > 0) {
        // Tail tile (unused for N = 4,194,304, kept for generality).
#pragma unroll
        for (int i = 0; i < 16; ++i) {
            int e = i * 8 + (tid >> 4);
            if ((long long)e < validE) {
                asm volatile("global_store_async_from_lds_b128 %0, %1, off"
                             :: "v"(gaddr), "v"(lds) : "memory");
            }
            lds   += 8u * (PITCH * 4);
            gaddr += (unsigned long long)(BLOCK * 16);
        }
    }
    // Drain outstanding async LDS->global transfers (s_endpgm also implies wait-idle).
    asm volatile("s_wait_asynccnt 0" ::: "memory");
}

extern "C" void kernel_launch(void* const* d_in, const int* in_sizes, int n_in,
                              void* d_out, int out_size, void* d_ws, size_t ws_size,
                              hipStream_t stream) {
    const float* len = (const float*)d_in[0];
    float* out = (float*)d_out;
    const int n = in_sizes[0];  // 4,194,304 edges
    const int blocks = (n + EDGES_PER_TILE - 1) / EDGES_PER_TILE;  // 32768
    edge_embed_kernel<<<blocks, BLOCK, 0, stream>>>(len, out, n);
}